// SetCriterionDETR_88648124991476
// MI455X (gfx1250) — compile-verified
//
#include <hip/hip_runtime.h>
#include <hip/hip_bf16.h>
#include <math.h>

#define B_   128
#define Q_   900
#define C_   256
#define NG_  50
#define EOS_COEF 0.1f
#define EPS_ 1e-6f

typedef __attribute__((ext_vector_type(2))) float v2f;
typedef __attribute__((ext_vector_type(8))) float v8f;

#if defined(__has_builtin)
#  if __has_builtin(__builtin_amdgcn_wmma_f32_16x16x4_f32)
#    define USE_WMMA_RED 1
#  endif
#endif
#ifndef USE_WMMA_RED
#  define USE_WMMA_RED 0
#endif

// ---------------------------------------------------------------- init ----
__global__ void k_init(int* __restrict__ tc, float* __restrict__ acc) {
  int i = blockIdx.x * blockDim.x + threadIdx.x;
  if (i < B_ * Q_) tc[i] = 0;
  if (i < 8) acc[i] = 0.0f;
}

// --------------------------------------------------------------- match ----
// One block per image. Stage pred xyxy + normalized GT in LDS, 50 parallel
// argmax-IoU reductions, then first-occurrence matching + L1/GIoU sums.
__global__ __launch_bounds__(256) void k_match(const float* __restrict__ pred,
                                               const float* __restrict__ gt,
                                               const int*   __restrict__ labels,
                                               int*   __restrict__ tc,
                                               float* __restrict__ g_acc) {
  __shared__ float s_pred[Q_ * 4];
  __shared__ float s_gt[NG_ * 4];
  __shared__ unsigned long long s_red[8];
  __shared__ int   s_qidx[NG_];
  __shared__ float s_wh[2];
  __shared__ float s_acc[3];

  const int b   = blockIdx.x;
  const int tid = threadIdx.x;
  const int wid = tid >> 5, lane = tid & 31;

  if (tid == 0) {
    float w = 1.0f, h = 1.0f;
    for (int g = 0; g < NG_; ++g) {
      w = fmaxf(w, gt[(size_t)(b * NG_ + g) * 4 + 2]);
      h = fmaxf(h, gt[(size_t)(b * NG_ + g) * 4 + 3]);
    }
    s_wh[0] = w; s_wh[1] = h;
    s_acc[0] = s_acc[1] = s_acc[2] = 0.0f;
  }
  __syncthreads();

  for (int t = tid; t < NG_ * 4; t += 256)
    s_gt[t] = gt[(size_t)b * NG_ * 4 + t] / s_wh[t & 1];

  for (int q = tid; q < Q_; q += 256) {
    const float4 pb = *(const float4*)(pred + (size_t)(b * Q_ + q) * 4);
    s_pred[q * 4 + 0] = pb.x - 0.5f * pb.z;
    s_pred[q * 4 + 1] = pb.y - 0.5f * pb.w;
    s_pred[q * 4 + 2] = pb.x + 0.5f * pb.z;
    s_pred[q * 4 + 3] = pb.y + 0.5f * pb.w;
  }
  __syncthreads();

  for (int g = 0; g < NG_; ++g) {
    const float gx1 = s_gt[g * 4 + 0], gy1 = s_gt[g * 4 + 1];
    const float gx2 = s_gt[g * 4 + 2], gy2 = s_gt[g * 4 + 3];
    const float ga = fmaxf(gx2 - gx1, 0.0f) * fmaxf(gy2 - gy1, 0.0f);
    unsigned long long best = 0ull;
    for (int q = tid; q < Q_; q += 256) {
      const float px1 = s_pred[q * 4 + 0], py1 = s_pred[q * 4 + 1];
      const float px2 = s_pred[q * 4 + 2], py2 = s_pred[q * 4 + 3];
      const float pa  = fmaxf(px2 - px1, 0.0f) * fmaxf(py2 - py1, 0.0f);
      const float ix1 = fmaxf(px1, gx1), iy1 = fmaxf(py1, gy1);
      const float ix2 = fminf(px2, gx2), iy2 = fminf(py2, gy2);
      const float inter = fmaxf(ix2 - ix1, 0.0f) * fmaxf(iy2 - iy1, 0.0f);
      const float uni = pa + ga - inter;
      const float iou = inter / (uni + EPS_);
      // iou >= 0 -> float bits order-preserving; tie-break = lowest q
      unsigned long long key =
          ((unsigned long long)__float_as_uint(iou) << 32) |
          (unsigned long long)(0xFFFFFFFFu - (unsigned)q);
      if (key > best) best = key;
    }
    for (int off = 16; off >= 1; off >>= 1) {
      unsigned long long o = __shfl_xor(best, off, 32);
      if (o > best) best = o;
    }
    if (lane == 0) s_red[wid] = best;
    __syncthreads();
    if (tid == 0) {
      unsigned long long m = s_red[0];
      for (int i = 1; i < 8; ++i) if (s_red[i] > m) m = s_red[i];
      s_qidx[g] = (int)(0xFFFFFFFFu - (unsigned)(m & 0xFFFFFFFFull));
    }
    __syncthreads();
  }

  if (tid < NG_) {
    const int g = tid;
    const int q = s_qidx[g];
    bool matched = true;
    for (int g2 = 0; g2 < g; ++g2)
      if (s_qidx[g2] == q) { matched = false; break; }
    if (matched) {
      tc[(size_t)b * Q_ + q] = labels[b * NG_ + g];
      const float px1 = s_pred[q * 4 + 0], py1 = s_pred[q * 4 + 1];
      const float px2 = s_pred[q * 4 + 2], py2 = s_pred[q * 4 + 3];
      const float gx1 = s_gt[g * 4 + 0], gy1 = s_gt[g * 4 + 1];
      const float gx2 = s_gt[g * 4 + 2], gy2 = s_gt[g * 4 + 3];
      const float l1 = fabsf(px1 - gx1) + fabsf(py1 - gy1) +
                       fabsf(px2 - gx2) + fabsf(py2 - gy2);
      const float pa = fmaxf(px2 - px1, 0.0f) * fmaxf(py2 - py1, 0.0f);
      const float ga = fmaxf(gx2 - gx1, 0.0f) * fmaxf(gy2 - gy1, 0.0f);
      const float ix1 = fmaxf(px1, gx1), iy1 = fmaxf(py1, gy1);
      const float ix2 = fminf(px2, gx2), iy2 = fminf(py2, gy2);
      const float inter = fmaxf(ix2 - ix1, 0.0f) * fmaxf(iy2 - iy1, 0.0f);
      const float uni = pa + ga - inter + EPS_;
      const float iou = inter / uni;
      const float cx1 = fminf(px1, gx1), cy1 = fminf(py1, gy1);
      const float cx2 = fmaxf(px2, gx2), cy2 = fmaxf(py2, gy2);
      const float areac = fmaxf(cx2 - cx1, 0.0f) * fmaxf(cy2 - cy1, 0.0f) + EPS_;
      const float giou = iou - (areac - uni) / areac;
      atomicAdd(&s_acc[0], l1);
      atomicAdd(&s_acc[1], 1.0f - giou);
      atomicAdd(&s_acc[2], 1.0f);
    }
  }
  __syncthreads();
  if (tid == 0) {
    atomicAdd(&g_acc[2], s_acc[0]);
    atomicAdd(&g_acc[3], s_acc[1]);
    atomicAdd(&g_acc[4], s_acc[2]);
  }
}

// ------------------------------------------------------------------ CE ----
// Each wave handles 16-row tiles (grid-stride). Tile staged to LDS,
// row-max via VALU, exp-sum via f32 WMMA against a ones matrix.
#define TS 260  // padded LDS row stride (dwords) to dodge bank conflicts

__global__ __launch_bounds__(64) void k_ce(const float* __restrict__ logits,
                                           const int*   __restrict__ tc,
                                           float* __restrict__ g_acc,
                                           int n_tiles, int tile_stride) {
  __shared__ float s_tile[2][16 * TS];
  __shared__ float s_rowmax[2][16];
  __shared__ float s_rowsum[2][16];

  const int tid  = threadIdx.x;
  const int w    = tid >> 5, lane = tid & 31;
  const int m    = lane & 15, half = lane >> 4;
  float* tile = s_tile[w];

  float acc_wnll = 0.0f, acc_w = 0.0f;

  for (int t = blockIdx.x * 2 + w; t < n_tiles; t += tile_stride) {
    const float* src = logits + (size_t)t * 16 * C_;
    for (int r = 0; r < 16; ++r) {
      const float4 v0 = *(const float4*)(src + r * C_ + lane * 4);
      const float4 v1 = *(const float4*)(src + r * C_ + 128 + lane * 4);
      *(float4*)&tile[r * TS + lane * 4]       = v0;
      *(float4*)&tile[r * TS + 128 + lane * 4] = v1;
    }
    __syncthreads();

    float mx = -1e30f;
    const float* rowp = &tile[m * TS + half * 128];
    for (int j = 0; j < 128; ++j) mx = fmaxf(mx, rowp[j]);
    mx = fmaxf(mx, __shfl_xor(mx, 16, 32));

#if USE_WMMA_RED
    // D[m,n] += sum_k A[m,k] * 1 : exact f32 row-sum on the matrix pipe.
    v8f d = {0.f, 0.f, 0.f, 0.f, 0.f, 0.f, 0.f, 0.f};
    v2f onesv = {1.0f, 1.0f};
    const float* basep = &tile[m * TS + half * 2];
    for (int kk = 0; kk < 64; ++kk) {
      v2f a;
      a.x = __expf(basep[kk * 4 + 0] - mx);
      a.y = __expf(basep[kk * 4 + 1] - mx);
      d = __builtin_amdgcn_wmma_f32_16x16x4_f32(
          false, a, false, onesv, (short)0, d, false, false);
    }
    // C/D layout: lanes 0-15 vgpr i = D[i, lane]; lanes 16-31 = D[8+i, ...]
    if (lane == 0)  for (int i = 0; i < 8; ++i) s_rowsum[w][i]     = d[i];
    if (lane == 16) for (int i = 0; i < 8; ++i) s_rowsum[w][8 + i] = d[i];
#else
    float s = 0.0f;
    for (int j = 0; j < 128; ++j) s += __expf(rowp[j] - mx);
    s += __shfl_xor(s, 16, 32);
    if (lane < 16) s_rowsum[w][lane] = s;
#endif
    if (lane < 16) s_rowmax[w][lane] = mx;
    __syncthreads();

    float wnll = 0.0f, wt = 0.0f;
    if (lane < 16) {
      const int row = t * 16 + lane;
      const int tci = tc[row];
      const float xt  = tile[lane * TS + tci];
      const float nll = __logf(s_rowsum[w][lane]) + s_rowmax[w][lane] - xt;
      wt   = (tci == 0) ? EOS_COEF : 1.0f;
      wnll = wt * nll;
    }
    acc_wnll += wnll;
    acc_w    += wt;
    __syncthreads();
  }

  for (int off = 16; off >= 1; off >>= 1) {
    acc_wnll += __shfl_xor(acc_wnll, off, 32);
    acc_w    += __shfl_xor(acc_w, off, 32);
  }
  if (lane == 0) {
    atomicAdd(&g_acc[0], acc_wnll);
    atomicAdd(&g_acc[1], acc_w);
  }
}

// ------------------------------------------------------------- finalize ---
__global__ void k_final(const float* __restrict__ acc, float* __restrict__ out) {
  if (threadIdx.x == 0 && blockIdx.x == 0) {
    out[0] = acc[0] / acc[1];
    const float n  = acc[4];
    const float ns = fmaxf(n, 1.0f);
    out[1] = (n > 0.0f) ? acc[2] / ns : 0.0f;
    out[2] = (n > 0.0f) ? acc[3] / ns : 0.0f;
  }
}

// ---------------------------------------------------------------------------
extern "C" void kernel_launch(void* const* d_in, const int* in_sizes, int n_in,
                              void* d_out, int out_size, void* d_ws, size_t ws_size,
                              hipStream_t stream) {
  (void)in_sizes; (void)n_in; (void)out_size; (void)ws_size;
  const float* logits = (const float*)d_in[0];   // [B,Q,C]
  const float* pred   = (const float*)d_in[1];   // [B,Q,4] cxcywh
  const float* gtb    = (const float*)d_in[2];   // [B,NG,4] xyxy px
  const int*   labels = (const int*)  d_in[3];   // [B,NG]
  float* out = (float*)d_out;                    // 3 scalars

  float* acc = (float*)d_ws;                     // 8 floats
  int*   tc  = (int*)((char*)d_ws + 64);         // B*Q ints

  k_init<<<(B_ * Q_ + 255) / 256, 256, 0, stream>>>(tc, acc);
  k_match<<<B_, 256, 0, stream>>>(pred, gtb, labels, tc, acc);
  // B*Q/16 = 7200 wave-tiles; 600 blocks x 2 waves -> stride 1200, 6 iters.
  k_ce<<<600, 64, 0, stream>>>(logits, tc, acc, (B_ * Q_) / 16, 1200);
  k_final<<<1, 1, 0, stream>>>(acc, out);
}